// GATNEModel_51969104281794
// MI455X (gfx1250) — compile-verified
//
#include <hip/hip_runtime.h>
#include <math.h>

typedef float v2f __attribute__((ext_vector_type(2)));
typedef float v8f __attribute__((ext_vector_type(8)));

#define EMB      200
#define EMB_U    20
#define DIM_A    20
#define USER_DIM 32
#define TEXT_DIM 768
#define SUM_PITCH (TEXT_DIM + 4)   // 772: 772 % 64 = 4 -> conflict-free A-fragment loads
#define NTYPES   2
#define S_NEIGH  10
#define NU_ROWS  50000
#define ROWS     16
#define THREADS  256

__global__ __launch_bounds__(THREADS)
void gatne_fused(const int* __restrict__ train_types,
                 const int* __restrict__ node_neigh,
                 const int* __restrict__ indices,
                 const float* __restrict__ user_features,
                 const float* __restrict__ text_features,
                 const float* __restrict__ neigh_features,
                 const float* __restrict__ text_embed_trans,
                 const float* __restrict__ user_embed_trans,
                 const float* __restrict__ tweet_u_trans,
                 const float* __restrict__ user_u_trans,
                 const float* __restrict__ trans_w,
                 const float* __restrict__ trans_s1,
                 const float* __restrict__ trans_s2,
                 float* __restrict__ out)
{
    __shared__ __align__(16) float s_sum[ROWS][SUM_PITCH];       // ~49.4 KB summed neighbor feats (current t)
    __shared__ float s_nteT[NTYPES][ROWS][EMB_U];                // tweet-projected type embed
    __shared__ float s_nteU[NTYPES][ROWS][EMB_U];                // user-projected type embed
    __shared__ float s_agg[ROWS][EMB_U];                         // attention-weighted embed
    __shared__ float s_red[ROWS][16];
    __shared__ float s_red2[ROWS][16];
    __shared__ float s_norm[ROWS];

    const int tid  = threadIdx.x;
    const int lane = tid & 31;
    const int wave = __builtin_amdgcn_readfirstlane(tid >> 5);   // force wave-uniform branches

    const int tr = tid >> 4;          // row team 0..15
    const int tl = tid & 15;          // lane in team

    const int grow = blockIdx.x * ROWS + tr;
    const int ty   = train_types[grow];
    const int is_tweet = (ty == 1);

    // ---------- Phase 1: base node embedding (kept in registers) ----------
    float base[13];
    #pragma unroll
    for (int j = 0; j < 13; ++j) base[j] = 0.f;
    {
        const int idx = indices[grow];
        const float* feat;
        const float* W;
        int K;
        if (idx < NU_ROWS) { feat = user_features + (long)idx * USER_DIM;              W = user_embed_trans; K = USER_DIM; }
        else               { feat = text_features + (long)(idx - NU_ROWS) * TEXT_DIM;  W = text_embed_trans; K = TEXT_DIM; }
        for (int k = 0; k < K; ++k) {
            const float f = feat[k];
            const float* Wk = W + (long)k * EMB + tl;
            #pragma unroll
            for (int j = 0; j < 13; ++j) {
                if (tl + 16 * j < EMB) base[j] += f * Wk[16 * j];
            }
        }
    }

    // ---------- Phase 2: per edge-type, gather-sum neighbors then WMMA-project ----------
    for (int t = 0; t < NTYPES; ++t) {
        // prefetch the weight matrix the WMMA phase will need (hidden under the gather)
        {
            const float* Wt = tweet_u_trans + (long)t * TEXT_DIM * EMB_U;
            for (int off = tid * 32; off < TEXT_DIM * EMB_U; off += THREADS * 32)
                __builtin_prefetch(Wt + off, 0, 3);
        }

        // zero the summed-feature tile
        for (int i = tid; i < ROWS * SUM_PITCH; i += THREADS)
            (&s_sum[0][0])[i] = 0.f;
        __syncthreads();

        // gather-sum: team tr owns row tr, lanes cover 64-float (256B) segments
        {
            const int nn_base = (grow * NTYPES + t) * S_NEIGH;
            int nidx[S_NEIGH];
            #pragma unroll
            for (int s = 0; s < S_NEIGH; ++s) nidx[s] = node_neigh[nn_base + s];
            const int kmax = is_tweet ? (TEXT_DIM / 64) : 1;   // user rows only need first 32 dims
            for (int k = 0; k < kmax; ++k) {
                const int f = k * 64 + tl * 4;
                float4 acc = make_float4(0.f, 0.f, 0.f, 0.f);
                #pragma unroll
                for (int s = 0; s < S_NEIGH; ++s) {
                    const float4 v = *(const float4*)(neigh_features + (long)nidx[s] * TEXT_DIM + f);
                    acc.x += v.x; acc.y += v.y; acc.z += v.z; acc.w += v.w;
                }
                *(float4*)(&s_sum[tr][f]) = acc;
            }
        }
        __syncthreads();

        // WMMA f32 16x16x4: waves 0,1 -> tweet proj (K=768), waves 2,3 -> user proj (K=32)
        // Lanes with u >= 20 compute a clamped-column garbage result that is never stored
        // (each D column depends only on the matching B column), so loads are unconditional.
        if (wave < 4) {
            const int ntile = wave & 1;
            const int m  = lane & 15;
            const int kh = lane >> 4;                      // which K half-pair this lane holds
            const int u  = ntile * 16 + m;                 // output column owned by this lane
            const int uc = (u < EMB_U) ? u : (EMB_U - 1);  // clamped column for loads

            if (wave < 2) {
                const float* W = tweet_u_trans + (long)t * TEXT_DIM * EMB_U;
                v8f c = {0.f, 0.f, 0.f, 0.f, 0.f, 0.f, 0.f, 0.f};
                #pragma unroll 8
                for (int k0 = 0; k0 < TEXT_DIM; k0 += 4) {
                    const int ka = k0 + kh * 2;
                    v2f a, b;
                    a.x = s_sum[m][ka];
                    a.y = s_sum[m][ka + 1];
                    b.x = W[ka * EMB_U + uc];
                    b.y = W[(ka + 1) * EMB_U + uc];
                    c = __builtin_amdgcn_wmma_f32_16x16x4_f32(false, a, false, b,
                                                              (short)0, c, false, false);
                }
                #pragma unroll
                for (int i = 0; i < 8; ++i) {
                    const int dm = i + kh * 8;             // D: VGPR i -> M = i + kh*8, N = u
                    if (u < EMB_U) s_nteT[t][dm][u] = c[i];
                }
            } else {
                const float* W = user_u_trans + (long)t * USER_DIM * EMB_U;
                v8f c = {0.f, 0.f, 0.f, 0.f, 0.f, 0.f, 0.f, 0.f};
                #pragma unroll
                for (int k0 = 0; k0 < USER_DIM; k0 += 4) {
                    const int ka = k0 + kh * 2;
                    v2f a, b;
                    a.x = s_sum[m][ka];
                    a.y = s_sum[m][ka + 1];
                    b.x = W[ka * EMB_U + uc];
                    b.y = W[(ka + 1) * EMB_U + uc];
                    c = __builtin_amdgcn_wmma_f32_16x16x4_f32(false, a, false, b,
                                                              (short)0, c, false, false);
                }
                #pragma unroll
                for (int i = 0; i < 8; ++i) {
                    const int dm = i + kh * 8;
                    if (u < EMB_U) s_nteU[t][dm][u] = c[i];
                }
            }
        }
        __syncthreads();
    }

    // ---------- Phase 3: 2-way attention ----------
    {
        const float* nt0 = is_tweet ? &s_nteT[0][tr][0] : &s_nteU[0][tr][0];
        const float* nt1 = is_tweet ? &s_nteT[1][tr][0] : &s_nteU[1][tr][0];
        const float* S1 = trans_s1 + (long)ty * EMB_U * DIM_A;
        const float* S2 = trans_s2 + (long)ty * DIM_A;
        float p0 = 0.f, p1 = 0.f;
        for (int aa = tl; aa < DIM_A; aa += 16) {
            float h0 = 0.f, h1 = 0.f;
            for (int uu = 0; uu < EMB_U; ++uu) {
                const float w = S1[uu * DIM_A + aa];
                h0 += nt0[uu] * w;
                h1 += nt1[uu] * w;
            }
            const float s2v = S2[aa];
            p0 += tanhf(h0) * s2v;
            p1 += tanhf(h1) * s2v;
        }
        s_red[tr][tl]  = p0;
        s_red2[tr][tl] = p1;
    }
    __syncthreads();
    if (tl == 0) {
        float s0 = 0.f, s1 = 0.f;
        #pragma unroll
        for (int i = 0; i < 16; ++i) { s0 += s_red[tr][i]; s1 += s_red2[tr][i]; }
        const float mx = fmaxf(s0, s1);
        const float e0 = expf(s0 - mx), e1 = expf(s1 - mx);
        const float inv = 1.f / (e0 + e1);
        const float a0 = e0 * inv, a1 = e1 * inv;
        const float* nt0 = is_tweet ? &s_nteT[0][tr][0] : &s_nteU[0][tr][0];
        const float* nt1 = is_tweet ? &s_nteT[1][tr][0] : &s_nteU[1][tr][0];
        for (int uu = 0; uu < EMB_U; ++uu)
            s_agg[tr][uu] = a0 * nt0[uu] + a1 * nt1[uu];
    }
    __syncthreads();

    // ---------- Phase 4: final 20x200 projection, add base, L2 normalize ----------
    {
        const float* Wf = trans_w + (long)ty * EMB_U * EMB;
        float ssq = 0.f;
        #pragma unroll
        for (int j = 0; j < 13; ++j) {
            const int e = tl + 16 * j;
            if (e < EMB) {
                float acc = base[j];
                for (int uu = 0; uu < EMB_U; ++uu)
                    acc += s_agg[tr][uu] * Wf[uu * EMB + e];
                base[j] = acc;
                ssq += acc * acc;
            }
        }
        s_red[tr][tl] = ssq;
    }
    __syncthreads();
    if (tl == 0) {
        float s = 0.f;
        #pragma unroll
        for (int i = 0; i < 16; ++i) s += s_red[tr][i];
        s_norm[tr] = fmaxf(sqrtf(s), 1e-12f);
    }
    __syncthreads();
    {
        const float inv = 1.f / s_norm[tr];
        float* orow = out + (long)grow * EMB;
        #pragma unroll
        for (int j = 0; j < 13; ++j) {
            const int e = tl + 16 * j;
            if (e < EMB) orow[e] = base[j] * inv;
        }
    }
}

extern "C" void kernel_launch(void* const* d_in, const int* in_sizes, int n_in,
                              void* d_out, int out_size, void* d_ws, size_t ws_size,
                              hipStream_t stream) {
    // setup_inputs order: 0 train_inputs (unused), 1 train_types, 2 node_neigh,
    // 3 indices, 4 user_features, 5 text_features, 6 neighbors_features,
    // 7 text_embed_trans, 8 user_embed_trans, 9 tweet_u_embed_trans,
    // 10 user_u_embed_trans, 11 trans_weights, 12 trans_weights_s1, 13 trans_weights_s2
    const int*   train_types = (const int*)d_in[1];
    const int*   node_neigh  = (const int*)d_in[2];
    const int*   indices     = (const int*)d_in[3];
    const float* user_feats  = (const float*)d_in[4];
    const float* text_feats  = (const float*)d_in[5];
    const float* neigh_feats = (const float*)d_in[6];
    const float* text_embed  = (const float*)d_in[7];
    const float* user_embed  = (const float*)d_in[8];
    const float* tweet_u     = (const float*)d_in[9];
    const float* user_u      = (const float*)d_in[10];
    const float* trans_w     = (const float*)d_in[11];
    const float* trans_s1    = (const float*)d_in[12];
    const float* trans_s2    = (const float*)d_in[13];

    const int B = in_sizes[1];          // 4096
    dim3 grid(B / ROWS), block(THREADS);
    hipLaunchKernelGGL(gatne_fused, grid, block, 0, stream,
                       train_types, node_neigh, indices,
                       user_feats, text_feats, neigh_feats,
                       text_embed, user_embed, tweet_u, user_u,
                       trans_w, trans_s1, trans_s2, (float*)d_out);
}